// QuantumLayer_31920196944607
// MI455X (gfx1250) — compile-verified
//
#include <hip/hip_runtime.h>
#include <stdint.h>

typedef __attribute__((ext_vector_type(2))) float v2f;
typedef __attribute__((ext_vector_type(8))) float v8f;

#if defined(__HIP_DEVICE_COMPILE__) && defined(__AMDGCN__)
#if !__has_builtin(__builtin_amdgcn_wmma_f32_16x16x4_f32)
#error "missing __builtin_amdgcn_wmma_f32_16x16x4_f32 (device pass)"
#endif
#endif

namespace {
constexpr int Dk     = 1024;       // K (feature dim)
constexpr int KB     = 64;         // K columns per staged tile
constexpr int NT     = Dk / KB;    // 16 tiles per row-block
constexpr int XSTR   = 68;         // padded LDS row stride (floats) -> conflict-free
constexpr int XBUF_F = 16 * XSTR;  // one 16-row tile buffer (1088 floats)
constexpr int WT_STR = 1028;       // padded stride for transposed weights
constexpr int WT_F   = 2 * WT_STR;
constexpr int WAVES  = 4;          // 128-thread block
constexpr int SMEM_F = WT_F + WAVES * 2 * XBUF_F;  // ~43 KB
}

struct c2 { float x, y; };

// Apply RX pair update: A' = c*A + (-i s)*B ; B' = (-i s)*A + c*B
__device__ __forceinline__ void rx_pair(float c, float s, c2 &A, c2 &B) {
  c2 a = A, b = B;
  A.x = c * a.x + s * b.y;
  A.y = c * a.y - s * b.x;
  B.x = c * b.x + s * a.y;
  B.y = c * b.y - s * a.x;
}

// Stage one 16-row x KB-col f32 tile: 8 async b128 ops (2 rows / instruction,
// 4 fully-consumed 128B lines per instruction).
__device__ __forceinline__ void issue_tile(uint64_t gbase, unsigned gOff, unsigned ldsOff) {
#pragma unroll
  for (int j = 0; j < 8; ++j) {
    asm volatile("global_load_async_to_lds_b128 %0, %1, %2"
                 :: "v"(ldsOff + (unsigned)(j * 2 * XSTR * 4)),
                    "v"(gOff + (unsigned)(j * 2 * Dk * 4)),
                    "s"(gbase)
                 : "memory");
  }
}

__global__ __launch_bounds__(WAVES * 32)
void qlayer_kernel(const float *__restrict__ x, const float *__restrict__ w,
                   const float *__restrict__ bias, const float *__restrict__ wq,
                   float *__restrict__ out) {
  __shared__ float smem[SMEM_F];
  const int tid = threadIdx.x;

  // Stage dense_w transposed: wT[q][k] at q*WT_STR + k
  for (int i = tid; i < 2 * Dk; i += WAVES * 32) {
    smem[(i & 1) * WT_STR + (i >> 1)] = w[i];
  }
  __syncthreads();

  const int wave = tid >> 5;
  const int lane = tid & 31;
  const int n = lane & 15;   // row-within-tile (B matrix N index)
  const int h = lane >> 4;   // lane half -> K sub-offset
  const int rowBase = ((int)blockIdx.x * WAVES + wave) * 16;

  const unsigned xbuf0_f = (unsigned)(WT_F + wave * 2 * XBUF_F);
  const unsigned smemLds = (unsigned)(uintptr_t)&smem[0];  // low 32 bits = LDS byte offset
  const unsigned dst0 = smemLds + (xbuf0_f + (unsigned)(h * XSTR + 4 * n)) * 4u;
  const unsigned dst1 = dst0 + (unsigned)XBUF_F * 4u;
  const unsigned gOff0 = ((unsigned)(rowBase + h) * (unsigned)Dk + 4u * (unsigned)n) * 4u;
  const uint64_t gbase = (uint64_t)(uintptr_t)x;

  v8f acc = {0.f, 0.f, 0.f, 0.f, 0.f, 0.f, 0.f, 0.f};

  issue_tile(gbase, gOff0, dst0);  // prologue: tile 0 -> buf0
  for (int t = 0; t < NT; ++t) {
    if (t + 1 < NT) {
      asm volatile("s_wait_dscnt 0" ::: "memory");       // buffer we overwrite was fully read
      issue_tile(gbase, gOff0 + (unsigned)((t + 1) * KB * 4),
                 ((t + 1) & 1) ? dst1 : dst0);
      asm volatile("s_wait_asynccnt 8" ::: "memory");    // tile t resident
    } else {
      asm volatile("s_wait_asynccnt 0" ::: "memory");
    }
    // B (x tile): lane -> x[row n, k0 + v + 2h]; A (wT): rows m>=2 are don't-care,
    // clamp to row 0 (same-address LDS broadcast, no conflicts).
    const unsigned bF = xbuf0_f + (unsigned)((t & 1) * XBUF_F) +
                        (unsigned)(n * XSTR + 2 * h);
    const unsigned mA = (n < 2) ? (unsigned)n : 0u;
    const unsigned aF = mA * (unsigned)WT_STR + (unsigned)(2 * h) + (unsigned)(t * KB);
#pragma unroll
    for (int s2 = 0; s2 < 16; ++s2) {
      v2f a = *(const v2f *)&smem[aF + 4 * s2];
      v2f b = *(const v2f *)&smem[bF + 4 * s2];
      acc = __builtin_amdgcn_wmma_f32_16x16x4_f32(false, a, false, b,
                                                  (short)0, acc, false, false);
    }
  }

  // D[m=qubit][n=row]: lane n (n<16) holds h0 in acc[0], h1 in acc[1].
  if (lane < 16) {
    float h0 = acc[0] + bias[0];
    float h1 = acc[1] + bias[1];
    float inv = rsqrtf(fmaxf(h0 * h0 + h1 * h1, 1e-12f));
    // Angles are bounded (|a|<=1 rad, weights ~0.05): hardware v_sin/v_cos are
    // plenty accurate here and avoid the libm slow-path code bloat.
    float a0 = 0.5f * h0 * inv, a1 = 0.5f * h1 * inv;
    float c0 = __cosf(a0), s0 = __sinf(a0);
    float c1 = __cosf(a1), s1 = __sinf(a1);
    // |00> after RX on q0 and RX on q1
    c2 S00 = {c0 * c1, 0.f};
    c2 S01 = {0.f, -c0 * s1};
    c2 S10 = {0.f, -s0 * c1};
    c2 S11 = {-s0 * s1, 0.f};
#pragma unroll
    for (int l = 0; l < 2; ++l) {
      float cw0 = __cosf(0.5f * wq[l * 2 + 0]);
      float sw0 = __sinf(0.5f * wq[l * 2 + 0]);
      float cw1 = __cosf(0.5f * wq[l * 2 + 1]);
      float sw1 = __sinf(0.5f * wq[l * 2 + 1]);
      rx_pair(cw0, sw0, S00, S10);  // RX on wire 0: pairs (00,10),(01,11)
      rx_pair(cw0, sw0, S01, S11);
      rx_pair(cw1, sw1, S00, S01);  // RX on wire 1: pairs (00,01),(10,11)
      rx_pair(cw1, sw1, S10, S11);
      c2 tmp = S10; S10 = S11; S11 = tmp;  // CNOT(0->1): swap |10>,|11>
    }
    float p00 = S00.x * S00.x + S00.y * S00.y;
    float p01 = S01.x * S01.x + S01.y * S01.y;
    float p10 = S10.x * S10.x + S10.y * S10.y;
    float p11 = S11.x * S11.x + S11.y * S11.y;
    v2f o = { (p00 + p01) - (p10 + p11), (p00 + p10) - (p01 + p11) };
    *(v2f *)&out[(size_t)(rowBase + n) * 2] = o;
  }
}

extern "C" void kernel_launch(void *const *d_in, const int *in_sizes, int n_in,
                              void *d_out, int out_size, void *d_ws, size_t ws_size,
                              hipStream_t stream) {
  (void)n_in; (void)out_size; (void)d_ws; (void)ws_size;
  const float *x  = (const float *)d_in[0];
  const float *w  = (const float *)d_in[1];
  const float *b  = (const float *)d_in[2];
  const float *wq = (const float *)d_in[3];
  float *out = (float *)d_out;
  const int Btot   = in_sizes[0] / Dk;        // 65536 rows
  const int blocks = Btot / (WAVES * 16);     // 1024 blocks of 128 threads
  qlayer_kernel<<<blocks, WAVES * 32, 0, stream>>>(x, w, b, wq, out);
}